// ChebConv_25391846654262
// MI455X (gfx1250) — compile-verified
//
#include <hip/hip_runtime.h>
#include <math.h>

// ---------------------------------------------------------------------------
// ChebConv (K=8) for MI455X / gfx1250.
//  - 8x [20000x512]@[512x512] fp32 GEMMs (~84 GFLOP) via V_WMMA_F32_16X16X4_F32
//  - A/B tiles double-buffered in LDS via GLOBAL_LOAD_ASYNC_TO_LDS_B128
//    (ASYNCcnt-tracked, overlapped with WMMA compute, one barrier/chunk)
//  - B fragments fetched as single ds_load_2addr_b32 K-pairs (no shuffles)
//  - 7 SPMM passes via L2-resident gathers + global_atomic_add_f32 scatters
// ---------------------------------------------------------------------------

typedef float v2f __attribute__((ext_vector_type(2)));
typedef float v8f __attribute__((ext_vector_type(8)));

#define CC 512      // in/out channels
#define BM 128      // block tile M
#define BN 64       // block tile N
#define BK 32       // K chunk staged in LDS
#define ASTRIDE 34  // padded LDS stride (floats) for A tile
#define BSTRIDE 80  // padded LDS stride (floats) for B tile (also 2addr offset1)

// ---- CDNA5 async global->LDS copy helpers (inline asm; no builtin exists) ----
// Per ISA 10/15.18: LDS dest address = LDS base + value of the VDST VGPR; GVS: mem = SGPR64 + VGPR32.
// Generic LDS pointer low 32 bits == wave-relative LDS byte address.
__device__ __forceinline__ unsigned lds_addr32(const void* p) {
    return (unsigned)(unsigned long long)p;
}
__device__ __forceinline__ void async_copy_b128(unsigned lds, unsigned voff,
                                                const void* sbase) {
    asm volatile("global_load_async_to_lds_b128 %0, %1, %2 offset:0"
                 :: "v"(lds), "v"(voff), "s"(sbase)
                 : "memory");
}
__device__ __forceinline__ void wait_asynccnt0() {
    asm volatile("s_wait_asynccnt 0x0" ::: "memory");
}
// One B fragment = {B[kb][n], B[kb+1][n]} = one ds_load_2addr_b32 with
// offset1 = BSTRIDE dwords; lands in an adjacent VGPR pair (no v_movs).
__device__ __forceinline__ v2f lds_load_kpair(const float* p) {
    v2f r;
    asm volatile("ds_load_2addr_b32 %0, %1 offset1:80"
                 : "=v"(r) : "v"(lds_addr32(p)));
    return r;
}

// -------------------------------- utility kernels --------------------------

__global__ void cheb_zero_kernel(float* __restrict__ p, int n4) {
    int i = blockIdx.x * blockDim.x + threadIdx.x;
    if (i < n4) ((float4*)p)[i] = make_float4(0.f, 0.f, 0.f, 0.f);
}

// dst = -src (float4 vectorized; src may alias dst for in-place negate)
__global__ void cheb_neg_kernel(const float* src, float* dst, int n4) {
    int i = blockIdx.x * blockDim.x + threadIdx.x;
    if (i < n4) {
        float4 v = ((const float4*)src)[i];
        ((float4*)dst)[i] = make_float4(-v.x, -v.y, -v.z, -v.w);
    }
}

__global__ void cheb_deg_kernel(const int* __restrict__ row, float* deg, int nE) {
    int e = blockIdx.x * blockDim.x + threadIdx.x;
    if (e < nE) atomicAdd(deg + row[e], 1.0f);
}

// in-place: deg[i] -> deg>0 ? rsqrt(deg) : 0
__global__ void cheb_dinv_kernel(float* deg, int n) {
    int i = blockIdx.x * blockDim.x + threadIdx.x;
    if (i < n) {
        float d = deg[i];
        deg[i] = (d > 0.0f) ? rsqrtf(d) : 0.0f;
    }
}

__global__ void cheb_lap_kernel(const int* __restrict__ row, const int* __restrict__ col,
                                const float* __restrict__ dinv, float* __restrict__ lap,
                                int nE) {
    int e = blockIdx.x * blockDim.x + threadIdx.x;
    if (e < nE) lap[e] = -dinv[row[e]] * dinv[col[e]];
}

// dst[row[e], :] += scale * lap[e] * src[col[e], :]
// 128 threads per edge, 4 channels per thread (float4 gather, 4 f32 atomics).
__global__ void cheb_spmm_kernel(const float* __restrict__ src, float* dst,
                                 const float* __restrict__ lap,
                                 const int* __restrict__ row, const int* __restrict__ col,
                                 float scale, int nE) {
    int t = blockIdx.x * blockDim.x + threadIdx.x;
    int e = t >> 7;          // edge id
    int g = t & 127;         // channel group (4 channels)
    if (e >= nE) return;
    int r = row[e];
    int c = col[e];
    float w = scale * lap[e];
    float4 v = *(const float4*)(src + (size_t)c * CC + g * 4);
    float* d = dst + (size_t)r * CC + g * 4;
    atomicAdd(d + 0, w * v.x);
    atomicAdd(d + 1, w * v.y);
    atomicAdd(d + 2, w * v.z);
    atomicAdd(d + 3, w * v.w);
}

// -------------------------------- WMMA GEMM --------------------------------
// out[M x 512] (+)= T[M x 512] @ W[512 x 512]   (fp32, V_WMMA_F32_16X16X4_F32)
//
// 256 threads = 8 waves. Block tile 128x64. Wave w computes the 16x64 strip
// at rows [w*16, w*16+16), i.e. four 16x16 C tiles (4 x v8f accumulators).
// Double-buffered LDS tiles staged with global_load_async_to_lds_b128;
// copies for chunk k+1 overlap the 32 WMMAs of chunk k; one barrier/chunk.
// OOB rows are clamped (branch-free) so EXEC stays all-ones; epilogue masks.
//
// Fragment layouts per CDNA5 ISA 7.12.2 (wave32):
//   A 16x4 : lane -> M = lane&15, holds K = 2*(lane>>4) + vgpr
//   B 4x16 : lane -> N = lane&15, holds K = 2*(lane>>4) + vgpr
//   C 16x16: vgpr v -> M = v + 8*(lane>>4), N = lane&15
__global__ __launch_bounds__(256)
void cheb_gemm_wmma(const float* __restrict__ T, const float* __restrict__ W,
                    const float* __restrict__ bias, float* __restrict__ out,
                    int M, int accumulate) {
    __shared__ float As[2][BM * ASTRIDE];
    __shared__ float Bs[2][BK * BSTRIDE];

    const int tid  = threadIdx.x;
    const int wave = tid >> 5;       // 0..7
    const int lane = tid & 31;
    const int half = lane >> 4;      // 0/1
    const int idx  = lane & 15;      // 0..15

    const int m0 = blockIdx.x * BM;
    const int n0 = blockIdx.y * BN;

    // Per-thread staging slots (fixed across K chunks).
    const int ar  = tid >> 3;              // A rows: ar, ar+32, ar+64, ar+96
    const int ac4 = (tid & 7) * 4;         // A col (float4)
    const int br  = tid >> 4;              // B rows: br, br+16
    const int bc4 = (tid & 15) * 4;        // B col (float4)

    // Issue the 6 async b128 copies staging chunk k0 into LDS buffer `buf`.
    auto stage = [&](int k0, int buf) {
        #pragma unroll
        for (int p = 0; p < 4; ++p) {
            int r  = p * 32 + ar;
            int gr = min(m0 + r, M - 1);             // clamp: EXEC stays all-ones
            unsigned voff = (unsigned)((size_t)gr * (CC * 4) + (size_t)(k0 + ac4) * 4);
            async_copy_b128(lds_addr32(&As[buf][r * ASTRIDE + ac4]), voff, T);
        }
        #pragma unroll
        for (int p = 0; p < 2; ++p) {
            int r = p * 16 + br;
            unsigned voff = (unsigned)((size_t)(k0 + r) * (CC * 4) + (size_t)(n0 + bc4) * 4);
            async_copy_b128(lds_addr32(&Bs[buf][r * BSTRIDE + bc4]), voff, W);
        }
    };

    v8f acc[4] = {};                 // four 16x16 f32 accumulators

    // Prologue: stage first chunk.
    stage(0, 0);
    wait_asynccnt0();
    __syncthreads();

    int cur = 0;
    for (int k0 = 0; k0 < CC; k0 += BK) {
        const int nxt = cur ^ 1;
        if (k0 + BK < CC) stage(k0 + BK, nxt);       // overlap with compute below

        const float* Arow = &As[cur][(wave * 16 + idx) * ASTRIDE];
        const float* Bbas = &Bs[cur][idx];
        #pragma unroll
        for (int kk = 0; kk < BK; kk += 4) {
            const int kb = kk + 2 * half;    // lane's K base (ISA A/B layout)
            v2f a;
            a.x = Arow[kb + 0];
            a.y = Arow[kb + 1];
            #pragma unroll
            for (int t = 0; t < 4; ++t) {
                v2f b = lds_load_kpair(Bbas + kb * BSTRIDE + t * 16);
                acc[t] = __builtin_amdgcn_wmma_f32_16x16x4_f32(
                    /*neg_a=*/false, a, /*neg_b=*/false, b,
                    /*c_mod=*/(short)0, acc[t],
                    /*reuse_a=*/false, /*reuse_b=*/false);
            }
        }

        wait_asynccnt0();            // next chunk's copies (issued above) landed
        __syncthreads();             // all waves done reading `cur`, writes visible
        cur = nxt;
    }

    // ---- epilogue (mask OOB rows here) ----
    #pragma unroll
    for (int t = 0; t < 4; ++t) {
        #pragma unroll
        for (int v = 0; v < 8; ++v) {
            int gr = m0 + wave * 16 + half * 8 + v;   // C-tile row mapping
            int gc = n0 + t * 16 + idx;
            if (gr < M) {
                float val = acc[t][v];
                float* p = out + (size_t)gr * CC + gc;
                if (accumulate) *p = *p + val;
                else            *p = val + bias[gc];
            }
        }
    }
}

// -------------------------------- launcher ---------------------------------

extern "C" void kernel_launch(void* const* d_in, const int* in_sizes, int n_in,
                              void* d_out, int out_size, void* d_ws, size_t ws_size,
                              hipStream_t stream) {
    const float* x      = (const float*)d_in[0];
    const int*   ei     = (const int*)d_in[1];
    const float* weight = (const float*)d_in[2];
    const float* bias   = (const float*)d_in[3];
    float*       out    = (float*)d_out;

    const int nE = in_sizes[1] / 2;
    const int M  = in_sizes[0] / CC;      // 20000 nodes
    const int Kw = in_sizes[2] / (CC * CC);
    const int* row = ei;
    const int* col = ei + nE;

    // workspace layout (floats): deg/dinv | lap | bufA | bufB
    float* deg  = (float*)d_ws;
    float* lap  = deg + ((M + 63) / 64) * 64;
    float* bufA = lap + ((nE + 63) / 64) * 64;
    float* bufB = bufA + (size_t)M * CC;

    const int TB = 256;
    const int n4T = M * CC / 4;   // float4 count of a node-feature buffer
    const int n4D = (M + 3) / 4;

    // ---- normalization: deg -> dinv -> lap ----
    cheb_zero_kernel<<<(n4D + TB - 1) / TB, TB, 0, stream>>>(deg, n4D);
    cheb_deg_kernel<<<(nE + TB - 1) / TB, TB, 0, stream>>>(row, deg, nE);
    cheb_dinv_kernel<<<(M + TB - 1) / TB, TB, 0, stream>>>(deg, M);
    cheb_lap_kernel<<<(nE + TB - 1) / TB, TB, 0, stream>>>(row, col, deg, lap, nE);

    // ---- Tx_1 = spmm(lap, x) -> bufA ----
    cheb_zero_kernel<<<(n4T + TB - 1) / TB, TB, 0, stream>>>(bufA, n4T);
    {
        long long nt = (long long)nE * 128;
        cheb_spmm_kernel<<<(unsigned)((nt + TB - 1) / TB), TB, 0, stream>>>(
            x, bufA, lap, row, col, 1.0f, nE);
    }

    // ---- out = x @ W0 + bias ; out += Tx_1 @ W1 ----
    dim3 ggrid((M + BM - 1) / BM, CC / BN);
    cheb_gemm_wmma<<<ggrid, TB, 0, stream>>>(x,    weight,                 bias,    out, M, 0);
    cheb_gemm_wmma<<<ggrid, TB, 0, stream>>>(bufA, weight + (size_t)CC*CC, nullptr, out, M, 1);

    // ---- Chebyshev recurrence: Tx_k = 2*spmm(Tx_{k-1}) - Tx_{k-2} ----
    // Parity rotation over two buffers: k even -> bufB, k odd -> bufA.
    for (int k = 2; k < Kw; ++k) {
        float*       dst  = (k % 2 == 0) ? bufB : bufA;  // Tx_k
        const float* src1 = (k % 2 == 0) ? bufA : bufB;  // Tx_{k-1}
        const float* src0 = (k == 2) ? x : dst;          // Tx_{k-2} (in-place for k>=3)

        cheb_neg_kernel<<<(n4T + TB - 1) / TB, TB, 0, stream>>>(src0, dst, n4T);
        long long nt = (long long)nE * 128;
        cheb_spmm_kernel<<<(unsigned)((nt + TB - 1) / TB), TB, 0, stream>>>(
            src1, dst, lap, row, col, 2.0f, nE);
        cheb_gemm_wmma<<<ggrid, TB, 0, stream>>>(
            dst, weight + (size_t)k * CC * CC, nullptr, out, M, 1);
    }
}